// AllTnn_4423816315096
// MI455X (gfx1250) — compile-verified
//
#include <hip/hip_runtime.h>
#include <hip/hip_bf16.h>
#include <math.h>

typedef float v2f __attribute__((ext_vector_type(2)));
typedef float v8f __attribute__((ext_vector_type(8)));

#define EPS 1e-5f

// ---------------------------------------------------------------------------
// Locally-connected conv (untied weights) + bias + ReLU, fused with the
// LayerNorm statistics reduction. One workgroup (256 threads, 8 wave32) per
// image; each thread owns HW/256 output pixels. Stats (mean, rstd) written to
// a small per-image array. Bandwidth-bound: every weight is touched once.
// ---------------------------------------------------------------------------
template <int H, int K, int PAD>
__launch_bounds__(256)
__global__ void lc_conv_relu_stats(const float* __restrict__ in,     // [B,H,H]
                                   const float* __restrict__ w,      // [H,H,K,K]
                                   const float* __restrict__ bias,   // [H,H]
                                   float* __restrict__ act,          // [B,H,H]
                                   float* __restrict__ stats) {      // [B,2]
    const int b   = blockIdx.x;
    const int tid = threadIdx.x;
    const int HW  = H * H;
    const float* img = in  + b * HW;
    float*       out = act + b * HW;

    float lsum = 0.f, lsq = 0.f;
    for (int p = tid; p < HW; p += 256) {
        const int i = p / H;
        const int j = p - i * H;
        const float* wp = w + p * (K * K);
        float s = bias[p];
#pragma unroll
        for (int ky = 0; ky < K; ++ky) {
            const int yy = i + ky - PAD;
            if (yy < 0 || yy >= H) continue;
            const float* row = img + yy * H;
#pragma unroll
            for (int kx = 0; kx < K; ++kx) {
                const int xx = j + kx - PAD;
                if (xx < 0 || xx >= H) continue;
                s = fmaf(row[xx], wp[ky * K + kx], s);
            }
        }
        s = fmaxf(s, 0.f);                 // ReLU
        out[p] = s;
        lsum += s;
        lsq  += s * s;
    }

    // Workgroup tree reduction in LDS (ds_load/ds_store path).
    __shared__ float ssum[256];
    __shared__ float ssq[256];
    ssum[tid] = lsum;
    ssq[tid]  = lsq;
    __syncthreads();
#pragma unroll
    for (int off = 128; off > 0; off >>= 1) {
        if (tid < off) {
            ssum[tid] += ssum[tid + off];
            ssq[tid]  += ssq[tid + off];
        }
        __syncthreads();
    }
    if (tid == 0) {
        const float inv = 1.f / (float)HW;
        const float mu  = ssum[0] * inv;
        const float var = ssq[0] * inv - mu * mu;
        stats[b * 2 + 0] = mu;
        stats[b * 2 + 1] = rsqrtf(var + EPS);
    }
}

// ---------------------------------------------------------------------------
// LayerNorm apply (affine g/beta) with optional fused 2x2 max-pool.
// ---------------------------------------------------------------------------
template <int H, bool POOL>
__launch_bounds__(256)
__global__ void ln_apply(const float* __restrict__ act,    // [B,H,H]
                         const float* __restrict__ g,      // [H,H]
                         const float* __restrict__ be,     // [H,H]
                         const float* __restrict__ stats,  // [B,2]
                         float* __restrict__ out) {        // [B,H,H] or [B,H/2,H/2]
    const int b  = blockIdx.x;
    const int HW = H * H;
    const float mu = stats[b * 2 + 0];
    const float rs = stats[b * 2 + 1];
    const float* a = act + b * HW;

    if (!POOL) {
        float* o = out + b * HW;
        for (int p = threadIdx.x; p < HW; p += 256)
            o[p] = (a[p] - mu) * rs * g[p] + be[p];
    } else {
        const int Ho  = H / 2;
        const int HWo = Ho * Ho;
        float* o = out + b * HWo;
        for (int p = threadIdx.x; p < HWo; p += 256) {
            const int i = p / Ho;
            const int j = p - i * Ho;
            float m = -INFINITY;
#pragma unroll
            for (int dy = 0; dy < 2; ++dy) {
#pragma unroll
                for (int dx = 0; dx < 2; ++dx) {
                    const int q = (2 * i + dy) * H + (2 * j + dx);
                    const float v = (a[q] - mu) * rs * g[q] + be[q];
                    m = fmaxf(m, v);
                }
            }
            o[p] = m;
        }
    }
}

// ---------------------------------------------------------------------------
// FC layer as a WMMA f32 GEMM: logits[64,1024] = h[64,256] @ fcw[1024,256]^T + fcb.
// One wave per 16x16 output tile; 4 M-tiles x 64 N-tiles = 256 waves.
// K stepped by 4 with V_WMMA_F32_16X16X4_F32.
//
// VGPR layouts per CDNA5 ISA 7.12.2:
//   A (16x4, M on lanes):  lanes 0-15 -> {K=k0,k0+1}, lanes 16-31 -> {K=k0+2,k0+3}
//   B (4x16, N on lanes):  mirrored (K split across half-waves, N = lane&15)
//   D (16x16): VGPR v -> M = m0 + v + (lane>=16 ? 8 : 0), N = n0 + (lane&15)
// ---------------------------------------------------------------------------
__launch_bounds__(256)
__global__ void fc_wmma_f32(const float* __restrict__ h,     // [64,256]
                            const float* __restrict__ fcw,   // [1024,256]
                            const float* __restrict__ fcb,   // [1024]
                            float* __restrict__ logits) {    // [64,1024]
    const int gwave = (blockIdx.x * blockDim.x + threadIdx.x) >> 5;  // 0..255
    const int lane  = threadIdx.x & 31;
    const int mt = gwave >> 6;     // 0..3
    const int nt = gwave & 63;     // 0..63
    const int m0 = mt * 16;
    const int n0 = nt * 16;

    const int idx15 = lane & 15;
    const int khalf = (lane >> 4) << 1;   // 0 for lanes 0-15, 2 for lanes 16-31

    const float* __restrict__ arow = h   + (m0 + idx15) * 256;  // A row (M on lanes)
    const float* __restrict__ brow = fcw + (n0 + idx15) * 256;  // B col = fcw row (N on lanes)

    v8f acc = {};
#pragma unroll 4
    for (int k0 = 0; k0 < 256; k0 += 4) {
        v2f a, bb;
        a.x  = arow[k0 + khalf];
        a.y  = arow[k0 + khalf + 1];
        bb.x = brow[k0 + khalf];
        bb.y = brow[k0 + khalf + 1];
        acc = __builtin_amdgcn_wmma_f32_16x16x4_f32(
            /*neg_a=*/false, a, /*neg_b=*/false, bb,
            /*c_mod=*/(short)0, acc, /*reuse_a=*/false, /*reuse_b=*/false);
    }

    const int col   = idx15;
    const int rbase = m0 + ((lane >> 4) << 3);
    const float bv  = fcb[n0 + col];
#pragma unroll
    for (int v = 0; v < 8; ++v)
        logits[(rbase + v) * 1024 + n0 + col] = acc[v] + bv;
}

// ---------------------------------------------------------------------------
// Row softmax: one workgroup per batch row of 1024 logits.
// ---------------------------------------------------------------------------
__launch_bounds__(256)
__global__ void softmax_rows(const float* __restrict__ logits,  // [64,1024]
                             float* __restrict__ out) {         // [64,1024]
    const int b = blockIdx.x;
    const float* r = logits + b * 1024;
    __shared__ float red[256];

    float m = -INFINITY;
    for (int i = threadIdx.x; i < 1024; i += 256) m = fmaxf(m, r[i]);
    red[threadIdx.x] = m;
    __syncthreads();
#pragma unroll
    for (int off = 128; off > 0; off >>= 1) {
        if (threadIdx.x < off)
            red[threadIdx.x] = fmaxf(red[threadIdx.x], red[threadIdx.x + off]);
        __syncthreads();
    }
    m = red[0];
    __syncthreads();

    float s = 0.f;
    for (int i = threadIdx.x; i < 1024; i += 256) s += __expf(r[i] - m);
    red[threadIdx.x] = s;
    __syncthreads();
#pragma unroll
    for (int off = 128; off > 0; off >>= 1) {
        if (threadIdx.x < off) red[threadIdx.x] += red[threadIdx.x + off];
        __syncthreads();
    }
    const float inv = 1.f / red[0];
    for (int i = threadIdx.x; i < 1024; i += 256)
        out[b * 1024 + i] = __expf(r[i] - m) * inv;
}

// ---------------------------------------------------------------------------
// Host launch. Input order: x, then (w,b,g,be) x 6, then fcw, fcb.
// Workspace: two ping-pong activation buffers (each 64*128*128 floats),
// LN stats (6 layers x 64 x 2), logits (64*1024).  ~8.7 MB total.
// ---------------------------------------------------------------------------
extern "C" void kernel_launch(void* const* d_in, const int* in_sizes, int n_in,
                              void* d_out, int out_size, void* d_ws, size_t ws_size,
                              hipStream_t stream) {
    const float* x = (const float*)d_in[0];
    const float* w[6];
    const float* bia[6];
    const float* g[6];
    const float* be[6];
    for (int i = 0; i < 6; ++i) {
        w[i]   = (const float*)d_in[1 + 4 * i + 0];
        bia[i] = (const float*)d_in[1 + 4 * i + 1];
        g[i]   = (const float*)d_in[1 + 4 * i + 2];
        be[i]  = (const float*)d_in[1 + 4 * i + 3];
    }
    const float* fcw = (const float*)d_in[25];
    const float* fcb = (const float*)d_in[26];
    float* out = (float*)d_out;

    const size_t BUF = (size_t)64 * 128 * 128;   // 1,048,576 floats
    float* bufA   = (float*)d_ws;
    float* bufB   = bufA + BUF;
    float* stats  = bufB + BUF;                  // 6 * 128 floats
    float* logits = stats + 6 * 128;             // 64 * 1024 floats

    const dim3 blk(256);
    const dim3 grdB(64);

    // L1: 128x128, k7 p3, LN, pool -> 64x64
    lc_conv_relu_stats<128, 7, 3><<<grdB, blk, 0, stream>>>(x,    w[0], bia[0], bufA, stats + 0 * 128);
    ln_apply<128, true>          <<<grdB, blk, 0, stream>>>(bufA, g[0], be[0], stats + 0 * 128, bufB);
    // L2: 64x64, k5 p2, LN
    lc_conv_relu_stats<64, 5, 2> <<<grdB, blk, 0, stream>>>(bufB, w[1], bia[1], bufA, stats + 1 * 128);
    ln_apply<64, false>          <<<grdB, blk, 0, stream>>>(bufA, g[1], be[1], stats + 1 * 128, bufB);
    // L3: 64x64, k5 p2, LN, pool -> 32x32
    lc_conv_relu_stats<64, 5, 2> <<<grdB, blk, 0, stream>>>(bufB, w[2], bia[2], bufA, stats + 2 * 128);
    ln_apply<64, true>           <<<grdB, blk, 0, stream>>>(bufA, g[2], be[2], stats + 2 * 128, bufB);
    // L4: 32x32, k3 p1, LN
    lc_conv_relu_stats<32, 3, 1> <<<grdB, blk, 0, stream>>>(bufB, w[3], bia[3], bufA, stats + 3 * 128);
    ln_apply<32, false>          <<<grdB, blk, 0, stream>>>(bufA, g[3], be[3], stats + 3 * 128, bufB);
    // L5: 32x32, k3 p1, LN, pool -> 16x16
    lc_conv_relu_stats<32, 3, 1> <<<grdB, blk, 0, stream>>>(bufB, w[4], bia[4], bufA, stats + 4 * 128);
    ln_apply<32, true>           <<<grdB, blk, 0, stream>>>(bufA, g[4], be[4], stats + 4 * 128, bufB);
    // L6: 16x16, k3 p1, LN  -> h [64, 256]
    lc_conv_relu_stats<16, 3, 1> <<<grdB, blk, 0, stream>>>(bufB, w[5], bia[5], bufA, stats + 5 * 128);
    ln_apply<16, false>          <<<grdB, blk, 0, stream>>>(bufA, g[5], be[5], stats + 5 * 128, bufB);

    // FC via WMMA f32: 256 waves = 8192 threads
    fc_wmma_f32<<<dim3(32), blk, 0, stream>>>(bufB, fcw, fcb, logits);

    // Softmax
    softmax_rows<<<grdB, blk, 0, stream>>>(logits, out);
}